// GCN_72035191489123
// MI455X (gfx1250) — compile-verified
//
#include <hip/hip_runtime.h>
#include <hip/hip_bf16.h>
#include <stdint.h>

// ---------------------------------------------------------------------------
// GCN 2-layer:  out = Agg( drop( Agg(x@W1)+b1 ) @ W2 ) + b2
// Agg(v)[c] = sum_e dinv[row]*dinv[col]*v[row]  (incl. self loops)
// ---------------------------------------------------------------------------

typedef float v2f __attribute__((ext_vector_type(2)));
typedef float v8f __attribute__((ext_vector_type(8)));

#define F_IN  512
#define F_H   16
#define F_OUT 40

// ---- CDNA5 async global->LDS (ASYNCcnt-tracked, no VGPR round trip) -------
__device__ __forceinline__ void async_ld_b128(uint32_t lds_off, const float* gaddr) {
  asm volatile("global_load_async_to_lds_b128 %0, %1, off"
               :: "v"(lds_off), "v"(gaddr) : "memory");
}
__device__ __forceinline__ void wait_async(void) {  // ASYNCcnt <= 4
  asm volatile("s_wait_asynccnt 0x4" ::: "memory");
}
__device__ __forceinline__ void wait_async0(void) { // ASYNCcnt == 0
  asm volatile("s_wait_asynccnt 0x0" ::: "memory");
}

// ------------------------------ degree / norm ------------------------------
__global__ void k_init_deg(float* __restrict__ deg, int n) {
  int i = blockIdx.x * blockDim.x + threadIdx.x;
  if (i < n) deg[i] = 1.0f;  // self loop contributes 1
}

__global__ void k_accum_deg(const int* __restrict__ col, float* __restrict__ deg, int E) {
  int e = blockIdx.x * blockDim.x + threadIdx.x;
  if (e < E) atomicAdd(&deg[col[e]], 1.0f);
}

__global__ void k_dinv(float* __restrict__ deg, int n) {
  int i = blockIdx.x * blockDim.x + threadIdx.x;
  if (i < n) {
    float d = deg[i];
    deg[i] = (d > 0.0f) ? rsqrtf(d) : 0.0f;
  }
}

// ------------------------------ GEMM1: xw = x @ W1  (N x 512 @ 512 x 16) ---
// 4 waves/block, one 16-row tile per wave. K streamed in 16 chunks of 32,
// double-buffered in LDS via async global->LDS loads; per-wave pipeline with
// no intra-loop barriers (ordering purely via ASYNCcnt).
__global__ __launch_bounds__(128) void k_gemm1(const float* __restrict__ x,
                                               const float* __restrict__ W1,
                                               float* __restrict__ xw, int n) {
  __shared__ float sW[F_IN * F_H];         // 32 KB: whole W1
  __shared__ float sA[4][2][16 * 36];      // per-wave double buffer, pad 36 (18 KB)

  const int t    = threadIdx.x;
  const int wave = t >> 5;
  const int lane = t & 31;
  const int m    = lane & 15;              // row within tile / col of B
  const int kh   = lane >> 4;              // K-half select

  // Load W1 into LDS (2048 float4, 128 threads x 16)
  for (int i = t; i < (F_IN * F_H) / 4; i += 128)
    ((float4*)sW)[i] = ((const float4*)W1)[i];
  __syncthreads();

  const int ntiles = n >> 4;
  int tile    = blockIdx.x * 4 + wave;
  bool active = tile < ntiles;
  int tcl     = active ? tile : (ntiles - 1);
  const int row0 = tcl << 4;

  const uint32_t lds0 = (uint32_t)(uintptr_t)&sA[wave][0][0];
  const uint32_t lds1 = (uint32_t)(uintptr_t)&sA[wave][1][0];

  // Stage one 16x32 chunk: 128 float4 -> 4 async b128 per lane
  const int r_st  = lane >> 3;             // 0..3  (row group base)
  const int c4_st = lane & 7;              // float4 within row
  auto stage = [&](uint32_t ldsbase, int k0) {
    for (int i = 0; i < 4; ++i) {
      int r = i * 4 + r_st;                // 0..15
      int rg = row0 + r; if (rg >= n) rg = n - 1;
      const float* src = x + (size_t)rg * F_IN + k0 + c4_st * 4;
      uint32_t dst = ldsbase + (uint32_t)(r * 36 + c4_st * 4) * 4u;
      async_ld_b128(dst, src);
    }
  };

  stage(lds0, 0);

  v8f c = {0.f, 0.f, 0.f, 0.f, 0.f, 0.f, 0.f, 0.f};

  for (int chunk = 0; chunk < 16; ++chunk) {
    const int buf = chunk & 1;
    if (chunk < 15) {
      stage(buf ? lds0 : lds1, (chunk + 1) * 32);  // prefetch next buffer
      wait_async();                                // first 4 (current buf) done
    } else {
      wait_async0();
    }
    const int k0 = chunk * 32;
    const float* __restrict__ sAb = &sA[wave][buf][0];
    for (int kk = 0; kk < 8; ++kk) {
      const int kb = kk * 4 + 2 * kh;
      v2f a, b;
      a.x = sAb[m * 36 + kb];
      a.y = sAb[m * 36 + kb + 1];
      b.x = sW[(k0 + kb) * F_H + m];
      b.y = sW[(k0 + kb + 1) * F_H + m];
      c = __builtin_amdgcn_wmma_f32_16x16x4_f32(false, a, false, b,
                                                (short)0, c, false, false);
    }
  }

  if (active) {
    for (int v = 0; v < 8; ++v) {
      int row = row0 + v + 8 * kh;
      if (row < n) xw[(size_t)row * F_H + m] = c[v];
    }
  }
}

// -------- h init: bias + self-loop term ------------------------------------
__global__ void k_init_h(const float* __restrict__ xw, const float* __restrict__ dinv,
                         const float* __restrict__ b1, float* __restrict__ h, int n) {
  int idx = blockIdx.x * blockDim.x + threadIdx.x;
  if (idx < n * F_H) {
    int i = idx >> 4, j = idx & 15;
    float di = dinv[i];
    h[idx] = b1[j] + di * di * xw[idx];
  }
}

// -------- edge aggregation layer 1: 4 lanes/edge, float4 gather ------------
__global__ void k_agg1(const int* __restrict__ row, const int* __restrict__ col,
                       const float* __restrict__ dinv, const float* __restrict__ xw,
                       float* __restrict__ h, int E) {
  int tid = blockIdx.x * blockDim.x + threadIdx.x;
  if (tid >= E * 4) return;
  int e = tid >> 2, g = (tid & 3) * 4;
  int r = row[e], cc = col[e];
  float nrm = dinv[r] * dinv[cc];
  float4 v = *(const float4*)(xw + (size_t)r * F_H + g);
  float* dst = h + (size_t)cc * F_H + g;
  atomicAdd(dst + 0, nrm * v.x);
  atomicAdd(dst + 1, nrm * v.y);
  atomicAdd(dst + 2, nrm * v.z);
  atomicAdd(dst + 3, nrm * v.w);
}

// -------- dropout (deterministic hash, p=0.5, scale 1/(1-p)=2) -------------
__device__ __forceinline__ uint32_t hash_u32(uint32_t x) {
  x ^= x >> 16; x *= 0x7feb352dU;
  x ^= x >> 15; x *= 0x846ca68bU;
  x ^= x >> 16; return x;
}

__global__ void k_dropout(float* __restrict__ h, int count) {
  int idx = blockIdx.x * blockDim.x + threadIdx.x;
  if (idx < count) {
    uint32_t r = hash_u32((uint32_t)idx * 2654435761u + 42u);
    h[idx] = (r & 1u) ? h[idx] * 2.0f : 0.0f;
  }
}

// ------------------------------ GEMM2: hw = h @ W2  (N x 16 @ 16 x 40) -----
// W2 zero-padded to 16x48 in LDS -> three 16-col WMMA tiles, K=16 (4 steps).
__global__ __launch_bounds__(128) void k_gemm2(const float* __restrict__ h,
                                               const float* __restrict__ W2,
                                               float* __restrict__ hw, int n) {
  __shared__ float sW2[16 * 48];
  __shared__ float sA[4][16 * 17];

  const int t    = threadIdx.x;
  const int wave = t >> 5;
  const int lane = t & 31;
  const int m    = lane & 15;
  const int kh   = lane >> 4;

  for (int i = t; i < 16 * 48; i += 128) {
    int k = i / 48, nn = i % 48;
    sW2[i] = (nn < F_OUT) ? W2[k * F_OUT + nn] : 0.0f;
  }

  const int ntiles = n >> 4;
  int tile    = blockIdx.x * 4 + wave;
  bool active = tile < ntiles;
  int tcl     = active ? tile : (ntiles - 1);
  const int row0 = tcl << 4;

  // Stage 16x16 A tile (64 float4, 2 per lane)
  for (int i = 0; i < 2; ++i) {
    int idx = i * 32 + lane;                 // 0..63
    int r   = idx >> 2;
    int c4  = idx & 3;
    int rg  = row0 + r; if (rg >= n) rg = n - 1;
    float4 v = *(const float4*)(h + (size_t)rg * F_H + c4 * 4);
    float* dst = &sA[wave][r * 17 + c4 * 4];
    dst[0] = v.x; dst[1] = v.y; dst[2] = v.z; dst[3] = v.w;
  }
  __syncthreads();

  for (int nt = 0; nt < 3; ++nt) {
    v8f c = {0.f, 0.f, 0.f, 0.f, 0.f, 0.f, 0.f, 0.f};
    for (int kk = 0; kk < 4; ++kk) {
      const int kb = kk * 4 + 2 * kh;
      v2f a, b;
      a.x = sA[wave][m * 17 + kb];
      a.y = sA[wave][m * 17 + kb + 1];
      b.x = sW2[kb * 48 + nt * 16 + m];
      b.y = sW2[(kb + 1) * 48 + nt * 16 + m];
      c = __builtin_amdgcn_wmma_f32_16x16x4_f32(false, a, false, b,
                                                (short)0, c, false, false);
    }
    int colo = nt * 16 + m;
    if (active && colo < F_OUT) {
      for (int v = 0; v < 8; ++v) {
        int row = row0 + v + 8 * kh;
        if (row < n) hw[(size_t)row * F_OUT + colo] = c[v];
      }
    }
  }
}

// -------- out init: bias + self-loop term ----------------------------------
__global__ void k_init_out(const float* __restrict__ hw, const float* __restrict__ dinv,
                           const float* __restrict__ b2, float* __restrict__ out, int n) {
  int idx = blockIdx.x * blockDim.x + threadIdx.x;
  if (idx < n * F_OUT) {
    int i = idx / F_OUT, j = idx % F_OUT;
    float di = dinv[i];
    out[idx] = b2[j] + di * di * hw[idx];
  }
}

// -------- edge aggregation layer 2: 10 float4-groups per edge --------------
__global__ void k_agg2(const int* __restrict__ row, const int* __restrict__ col,
                       const float* __restrict__ dinv, const float* __restrict__ hw,
                       float* __restrict__ out, int E) {
  int tid = blockIdx.x * blockDim.x + threadIdx.x;
  if (tid >= E * 10) return;
  int e = tid / 10, g = (tid % 10) * 4;
  int r = row[e], cc = col[e];
  float nrm = dinv[r] * dinv[cc];
  float4 v = *(const float4*)(hw + (size_t)r * F_OUT + g);
  float* dst = out + (size_t)cc * F_OUT + g;
  atomicAdd(dst + 0, nrm * v.x);
  atomicAdd(dst + 1, nrm * v.y);
  atomicAdd(dst + 2, nrm * v.z);
  atomicAdd(dst + 3, nrm * v.w);
}

// ---------------------------------------------------------------------------
extern "C" void kernel_launch(void* const* d_in, const int* in_sizes, int n_in,
                              void* d_out, int out_size, void* d_ws, size_t ws_size,
                              hipStream_t stream) {
  const float* x  = (const float*)d_in[0];
  const int*   ei = (const int*)d_in[1];
  const float* W1 = (const float*)d_in[2];
  const float* b1 = (const float*)d_in[3];
  const float* W2 = (const float*)d_in[4];
  const float* b2 = (const float*)d_in[5];
  float* out = (float*)d_out;

  const int n = in_sizes[0] / F_IN;      // 100000
  const int E = in_sizes[1] / 2;         // 3200000
  const int* row = ei;
  const int* col = ei + E;

  // workspace carve: dinv[n] | xw[n*16] | h[n*16] | hw[n*40]
  float* dinv = (float*)d_ws;
  float* xw   = dinv + n;
  float* h    = xw + (size_t)n * F_H;
  float* hw   = h + (size_t)n * F_H;

  const int B = 256;
  const int ntiles = (n + 15) / 16;
  const int gblocks = (ntiles + 3) / 4;

  k_init_deg<<<(n + B - 1) / B, B, 0, stream>>>(dinv, n);
  k_accum_deg<<<(E + B - 1) / B, B, 0, stream>>>(col, dinv, E);
  k_dinv<<<(n + B - 1) / B, B, 0, stream>>>(dinv, n);

  k_gemm1<<<gblocks, 128, 0, stream>>>(x, W1, xw, n);

  k_init_h<<<((n * F_H) + B - 1) / B, B, 0, stream>>>(xw, dinv, b1, h, n);
  k_agg1<<<((E * 4) + B - 1) / B, B, 0, stream>>>(row, col, dinv, xw, h, E);
  k_dropout<<<((n * F_H) + B - 1) / B, B, 0, stream>>>(h, n * F_H);

  k_gemm2<<<gblocks, 128, 0, stream>>>(h, W2, hw, n);

  k_init_out<<<((n * F_OUT) + B - 1) / B, B, 0, stream>>>(hw, dinv, b2, out, n);
  k_agg2<<<((E * 10) + B - 1) / B, B, 0, stream>>>(row, col, dinv, hw, out, E);
}